// SpatialAttentionGNNLayer_38912403701955
// MI455X (gfx1250) — compile-verified
//
#include <hip/hip_runtime.h>

typedef __attribute__((ext_vector_type(2))) float v2f;
typedef __attribute__((ext_vector_type(8))) float v8f;
typedef unsigned int u32x4 __attribute__((ext_vector_type(4)));
typedef int i32x8 __attribute__((ext_vector_type(8)));
typedef int i32x4 __attribute__((ext_vector_type(4)));

#define GAT_ALPHA 0.2f
#define BATCH 8
#define NNODE 2048
#define FDIM 128

__device__ __forceinline__ float leaky(float v) { return v > 0.f ? v : GAT_ALPHA * v; }

// ---------------------------------------------------------------------------
// TDM 2D tile load: global -> LDS via tensor_load_to_lds (TENSORcnt-tracked).
// g1 carries (data_size=4B, tensor_dim0/1, tile_dim0/1, dim0 stride) packed
// per D# group-1 layout; group0 built here from the two addresses.
// 6-arg builtin form (clang-23 / therock headers).
// ---------------------------------------------------------------------------
__device__ __forceinline__ void tdm_load_2d(const float* gp, const float* lp, i32x8 g1) {
  const unsigned long long ga = (unsigned long long)(uintptr_t)gp;
  const unsigned int la = (unsigned int)(uintptr_t)lp;  // low 32 bits = LDS byte offset
  u32x4 g0;
  g0.x = 1u;                                                  // count=1, user mode
  g0.y = la;                                                  // lds_addr [63:32]
  g0.z = (unsigned int)ga;                                    // global_addr [95:64]
  g0.w = (unsigned int)((ga >> 32) & 0x01FFFFFFull) | 0x80000000u;  // addr[56:32] | type=2
  i32x4 z4 = (i32x4)0;
  i32x8 z8 = (i32x8)0;
  __builtin_amdgcn_tensor_load_to_lds(g0, g1, z4, z4, z8, 0);
}

// D# group-1 constants: dw0 = data_size(4B)<<16; dw1 = tensor_dim0<<16;
// dw2 = tensor_dim1<<16; dw3 = tile_dim0<<16; dw4 = tile_dim1; dw5 = dim0_stride.
#define G1_TILE(td0, td1, t0, t1, stride) \
  { 0x00020000, (int)((td0) << 16), (int)((td1) << 16), (int)((t0) << 16), (int)(t1), (int)(stride), 0, 0 }

// -------------------------------------------------------------------------
// Kernel 1: h = x @ W   ([B*N, 128] @ [128, 128]) via v_wmma_f32_16x16x4_f32
// Block = 128 threads (4 waves); each wave computes a 16x128 tile of h.
// x tile staged to LDS by TDM; W operand reads are L2-broadcast.
// -------------------------------------------------------------------------
__global__ __launch_bounds__(128) void gat_gemm_h(const float* __restrict__ x,
                                                  const float* __restrict__ W,
                                                  float* __restrict__ h) {
  __shared__ float xs[4][16 * FDIM];  // 32 KB
  const int w    = threadIdx.x >> 5;
  const int lane = threadIdx.x & 31;
  const int rb   = blockIdx.x * 64 + w * 16;  // flattened (b*N + n) row base
  const int hi   = lane >> 4;
  const int ln   = lane & 15;

  // TDM: stage 16 rows x 128 cols of x into this wave's LDS slice
  const i32x8 g1_x = G1_TILE(128, 16, 128, 16, 128);
  tdm_load_2d(x + (size_t)rb * FDIM, &xs[w][0], g1_x);
  __builtin_amdgcn_s_wait_tensorcnt(0);

  v8f acc[8];
#pragma unroll
  for (int t = 0; t < 8; ++t) acc[t] = (v8f)0.0f;

#pragma unroll 4
  for (int s = 0; s < 32; ++s) {           // K = 128 in steps of 4
    const int k0 = 4 * s + 2 * hi;
    v2f av;                                 // A: M = ln, K = k0 + {0,1}
    av.x = xs[w][ln * FDIM + k0];
    av.y = xs[w][ln * FDIM + k0 + 1];
#pragma unroll
    for (int t = 0; t < 8; ++t) {           // B: N = ln, K = k0 + {0,1}
      v2f bv;
      bv.x = W[(k0)     * FDIM + t * 16 + ln];
      bv.y = W[(k0 + 1) * FDIM + t * 16 + ln];
      acc[t] = __builtin_amdgcn_wmma_f32_16x16x4_f32(false, av, false, bv,
                                                     (short)0, acc[t], false, false);
    }
  }

#pragma unroll
  for (int t = 0; t < 8; ++t)
#pragma unroll
    for (int r = 0; r < 8; ++r)             // C/D: M = r + 8*hi, N = ln
      h[(size_t)(rb + r + 8 * hi) * FDIM + t * 16 + ln] = acc[t][r];
}

// -------------------------------------------------------------------------
// Kernel 2: e_src[row] = h[row,:]·a_src, e_dst[row] = h[row,:]·a_dst
// -------------------------------------------------------------------------
__global__ __launch_bounds__(128) void gat_edge_terms(const float* __restrict__ h,
                                                      const float* __restrict__ a_src,
                                                      const float* __restrict__ a_dst,
                                                      float* __restrict__ e_src,
                                                      float* __restrict__ e_dst) {
  const int w = threadIdx.x >> 5, lane = threadIdx.x & 31;
  const int row = blockIdx.x * 4 + w;
  float4 hv = ((const float4*)(h + (size_t)row * FDIM))[lane];
  float4 as = ((const float4*)a_src)[lane];
  float4 ad = ((const float4*)a_dst)[lane];
  float ssum = hv.x * as.x + hv.y * as.y + hv.z * as.z + hv.w * as.w;
  float dsum = hv.x * ad.x + hv.y * ad.y + hv.z * ad.z + hv.w * ad.w;
#pragma unroll
  for (int off = 16; off > 0; off >>= 1) {
    ssum += __shfl_xor(ssum, off, 32);
    dsum += __shfl_xor(dsum, off, 32);
  }
  if (lane == 0) { e_src[row] = ssum; e_dst[row] = dsum; }
}

// -------------------------------------------------------------------------
// Kernel 2b: gmax[b] = max_j e_dst[b, j]  (exp-overflow bound for softmax)
// -------------------------------------------------------------------------
__global__ __launch_bounds__(256) void gat_gmax(const float* __restrict__ e_dst,
                                                float* __restrict__ gmax) {
  __shared__ float red[256];
  const int b = blockIdx.x;
  float m = -3.0e38f;
  for (int j = threadIdx.x; j < NNODE; j += 256) m = fmaxf(m, e_dst[b * NNODE + j]);
  red[threadIdx.x] = m;
  __syncthreads();
  for (int s = 128; s > 0; s >>= 1) {
    if (threadIdx.x < s) red[threadIdx.x] = fmaxf(red[threadIdx.x], red[threadIdx.x + s]);
    __syncthreads();
  }
  if (threadIdx.x == 0) gmax[b] = red[0];
}

// -------------------------------------------------------------------------
// Kernel 3: fused masked-softmax attention + ELU.
// grid = (N/64, B); block = 128 threads = 4 waves, each owns a 16-row i-tile.
// h[j-tile] double-buffered in LDS via TDM; adj tile single-buffered via TDM.
// One-pass softmax: m_i = leaky(e_src_i + max_j e_dst_j) >= all row logits.
// -------------------------------------------------------------------------
__global__ __launch_bounds__(128) void gat_attn(const float* __restrict__ h,
                                                const float* __restrict__ adj,
                                                const float* __restrict__ e_src,
                                                const float* __restrict__ e_dst,
                                                const float* __restrict__ gmax,
                                                float* __restrict__ out) {
  __shared__ float h_s[2][16 * FDIM];  // 16 KB double-buffered h[j-tile,:]
  __shared__ float adj_s[64 * 16];     //  4 KB adj[block i-rows, j-tile]
  __shared__ float eds[NNODE];         //  8 KB e_dst[b,:]

  const int b   = blockIdx.y;
  const int ib0 = blockIdx.x * 64;
  const int w = threadIdx.x >> 5, lane = threadIdx.x & 31;
  const int hi = lane >> 4, ln = lane & 15;
  const int iw = ib0 + w * 16;  // wave's i-tile base
  const int i  = iw + ln;       // this lane's P-matrix row (node id)
  const int il = w * 16 + ln;   // row within block's 64 adj rows

  // TDM descriptors: h slice = 128 x 4 rows (stride 128); adj slice = 16 x 16 rows (stride 2048)
  const i32x8 g1_h   = G1_TILE(128, 4, 128, 4, 128);
  const i32x8 g1_adj = G1_TILE(16, 16, 16, 16, NNODE);

  {  // preload e_dst[b,:] into LDS
    const float4* src = (const float4*)(e_dst + b * NNODE);
    float4* dst = (float4*)eds;
    for (int q = threadIdx.x; q < NNODE / 4; q += 128) dst[q] = src[q];
  }
  const float es_i = e_src[b * NNODE + i];
  const float m_i  = leaky(es_i + gmax[b]);

  float srow = 0.f;
  v8f acc[8];
#pragma unroll
  for (int t = 0; t < 8; ++t) acc[t] = (v8f)0.0f;

  // prologue: stage h rows for j0 = 0 (each wave its 4-row slice)
  tdm_load_2d(h + (size_t)(b * NNODE + 4 * w) * FDIM, &h_s[0][4 * w * FDIM], g1_h);

  for (int j0 = 0; j0 < NNODE; j0 += 16) {
    const int cur = (j0 >> 4) & 1;
    const int nxt = cur ^ 1;
    // issue adj(k) then h(k+1); in-order TENSORcnt completion means
    // wait(<=1) guarantees h(k) and adj(k) are done, h(k+1) may be in flight
    tdm_load_2d(adj + (size_t)(ib0 + 16 * w) * NNODE + j0, &adj_s[16 * w * 16], g1_adj);
    if (j0 + 16 < NNODE) {
      tdm_load_2d(h + (size_t)(b * NNODE + j0 + 16 + 4 * w) * FDIM,
                  &h_s[nxt][4 * w * FDIM], g1_h);
      __builtin_amdgcn_s_wait_tensorcnt(1);
    } else {
      __builtin_amdgcn_s_wait_tensorcnt(0);
    }
    __syncthreads();  // all waves' slices for step k are in LDS

    // build P in the WMMA A layout: lane holds M = ln, K = 4s + 2*hi + v
    float p[4][2];
#pragma unroll
    for (int s = 0; s < 4; ++s) {
#pragma unroll
      for (int v = 0; v < 2; ++v) {
        const int c = 4 * s + 2 * hi + v;
        const int j = j0 + c;
        const float a = adj_s[il * 16 + c];
        float pv = 0.f;
        if ((a > 0.f) || (j == i))
          pv = __expf(leaky(es_i + eds[j]) - m_i);
        p[s][v] = pv;
        srow += pv;
      }
    }
#pragma unroll
    for (int s = 0; s < 4; ++s) {
      v2f av; av.x = p[s][0]; av.y = p[s][1];
      const int k0 = 4 * s + 2 * hi;
#pragma unroll
      for (int t = 0; t < 8; ++t) {  // B: N = ln, K = k0 + {0,1} from LDS h tile
        v2f bv;
        bv.x = h_s[cur][(k0)     * FDIM + t * 16 + ln];
        bv.y = h_s[cur][(k0 + 1) * FDIM + t * 16 + ln];
        acc[t] = __builtin_amdgcn_wmma_f32_16x16x4_f32(false, av, false, bv,
                                                       (short)0, acc[t], false, false);
      }
    }
    __syncthreads();  // everyone done reading buf[cur]/adj_s before they are rewritten
  }

  // row softmax denominators: lane halves cover disjoint j sets -> xor-16 combine
  const float S = srow + __shfl_xor(srow, 16, 32);
#pragma unroll
  for (int r = 0; r < 8; ++r) {
    const float Sr = __shfl(S, r + 8 * hi, 32);  // S for acc row M = r + 8*hi
    const float inv = 1.0f / Sr;
    const int orow = iw + r + 8 * hi;
#pragma unroll
    for (int t = 0; t < 8; ++t) {
      float v = acc[t][r] * inv;
      v = v > 0.f ? v : (__expf(v) - 1.0f);  // ELU (alpha=1)
      out[(size_t)(b * NNODE + orow) * FDIM + t * 16 + ln] = v;
    }
  }
}

// -------------------------------------------------------------------------
extern "C" void kernel_launch(void* const* d_in, const int* in_sizes, int n_in,
                              void* d_out, int out_size, void* d_ws, size_t ws_size,
                              hipStream_t stream) {
  const float* x     = (const float*)d_in[0];
  const float* adj   = (const float*)d_in[1];
  const float* W     = (const float*)d_in[2];
  const float* a_src = (const float*)d_in[3];
  const float* a_dst = (const float*)d_in[4];
  float* out = (float*)d_out;

  float* h     = (float*)d_ws;                        // 16384*128 f32 = 8 MB
  float* e_src = h + (size_t)BATCH * NNODE * FDIM;    // 16384
  float* e_dst = e_src + (size_t)BATCH * NNODE;       // 16384
  float* gmax  = e_dst + (size_t)BATCH * NNODE;       // 8

  gat_gemm_h<<<dim3((BATCH * NNODE) / 64), dim3(128), 0, stream>>>(x, W, h);
  gat_edge_terms<<<dim3((BATCH * NNODE) / 4), dim3(128), 0, stream>>>(h, a_src, a_dst, e_src, e_dst);
  gat_gmax<<<dim3(BATCH), dim3(256), 0, stream>>>(e_dst, gmax);
  gat_attn<<<dim3(NNODE / 64, BATCH), dim3(128), 0, stream>>>(h, adj, e_src, e_dst, gmax, out);
}